// FastCachedSddmmLinear_47579647705405
// MI455X (gfx1250) — compile-verified
//
#include <hip/hip_runtime.h>
#include <hip/hip_bf16.h>

// ---------------------------------------------------------------------------
// FastCachedSddmmLinear for MI455X (gfx1250)
//
// Strategy: the gathered GEMV over a random half of the columns touches ~every
// cacheline of W anyway, so we convert it to a dense masked GEMV:
//     s[j] = multiplicity(j) * x[j]   (0/1 for topk set; counts for cached)
//     out  = W @ s + bias
// Streaming W = 180 MB -> ~7.7us at 23.3 TB/s; compute (45 MFLOP) is free, so
// we keep full f32 precision and use V_WMMA_F32_16X16X4_F32 for the GEMV.
// ---------------------------------------------------------------------------

#define IN_F   11008
#define OUT_F  4096
#define TOPK   5504
#define NCACHE 64
#define MW     344        // IN_F / 32 bit-mask words
#define NCHUNK 32         // K-split of GEMV
#define CHUNK  344        // IN_F / NCHUNK (divisible by 4)
#define SEG    11         // ceil(IN_F / 1024) per-thread segment in select

typedef __attribute__((ext_vector_type(2))) float v2f;
typedef __attribute__((ext_vector_type(8))) float v8f;

// ---------------------------------------------------------------------------
// 1) Exact top-5504 of |x| via 4-pass byte radix-select, single workgroup.
//    Produces an 11008-bit mask. Ties broken by ascending index (matches
//    the set semantics of the reference; order is irrelevant downstream).
// ---------------------------------------------------------------------------
__global__ __launch_bounds__(1024) void select_topk_kernel(
    const float* __restrict__ x, unsigned* __restrict__ maskbits_g) {
  __shared__ unsigned skey[IN_F];
  __shared__ unsigned hist[256];
  __shared__ unsigned scn[1024];
  __shared__ unsigned mbits[MW];
  __shared__ unsigned sh_prefix, sh_k;

  const int tid = threadIdx.x;

  for (int j = tid; j < IN_F; j += 1024)
    skey[j] = __float_as_uint(x[j]) & 0x7fffffffu;   // |x| bits (monotone)
  for (int w = tid; w < MW; w += 1024) mbits[w] = 0;
  if (tid == 0) { sh_prefix = 0u; sh_k = TOPK; }
  __syncthreads();

  // radix select, high byte -> low byte
  for (int p = 3; p >= 0; --p) {
    for (int b = tid; b < 256; b += 1024) hist[b] = 0;
    __syncthreads();
    const unsigned prefix = sh_prefix;
    for (int j = tid; j < IN_F; j += 1024) {
      unsigned key = skey[j];
      int sh = (p + 1) * 8;
      bool match = (p == 3) || ((key >> sh) == (prefix >> sh));
      if (match) atomicAdd(&hist[(key >> (p * 8)) & 255u], 1u);
    }
    __syncthreads();
    if (tid == 0) {
      unsigned k = sh_k, cum = 0; int sel = 0;
      for (int b = 255; b >= 0; --b) {
        unsigned h = hist[b];
        if (cum + h >= k) { sel = b; sh_k = k - cum; break; }
        cum += h;
      }
      sh_prefix = prefix | ((unsigned)sel << (p * 8));
    }
    __syncthreads();
  }

  const unsigned T = sh_prefix;        // key value of the 5504-th largest
  const unsigned need_eq = sh_k;       // how many ==T elements to take

  // exact rank of ==T elements by index: per-thread contiguous segments + scan
  const int start = tid * SEG;
  const int end   = min(start + SEG, IN_F);
  unsigned cnt = 0;
  for (int j = start; j < end; ++j) if (skey[j] == T) cnt++;
  scn[tid] = cnt;
  __syncthreads();
  for (int off = 1; off < 1024; off <<= 1) {
    unsigned v = (tid >= off) ? scn[tid - off] : 0u;
    __syncthreads();
    scn[tid] += v;
    __syncthreads();
  }
  unsigned rank = scn[tid] - cnt;      // exclusive prefix of ==T before start

  for (int j = start; j < end; ++j) {
    unsigned key = skey[j];
    bool sel = false;
    if (key > T) sel = true;
    else if (key == T) { if (rank < need_eq) sel = true; rank++; }
    if (sel) atomicOr(&mbits[j >> 5], 1u << (j & 31));
  }
  __syncthreads();
  for (int w = tid; w < MW; w += 1024) maskbits_g[w] = mbits[w];
}

// ---------------------------------------------------------------------------
// 2) Per-cache intersection with the top-k set (dedup via LDS bitmask).
// ---------------------------------------------------------------------------
__global__ __launch_bounds__(256) void recall_kernel(
    const long long* __restrict__ cached, const unsigned* __restrict__ maskbits,
    unsigned* __restrict__ inter) {
  __shared__ unsigned bits[MW];
  __shared__ unsigned red[256];
  const int i = blockIdx.x, tid = threadIdx.x;
  for (int w = tid; w < MW; w += 256) bits[w] = 0;
  __syncthreads();
  for (int t = tid; t < TOPK; t += 256) {
    unsigned idx = (unsigned)cached[(size_t)i * TOPK + t];
    atomicOr(&bits[idx >> 5], 1u << (idx & 31));
  }
  __syncthreads();
  unsigned c = 0;
  for (int w = tid; w < MW; w += 256) c += __popc(bits[w] & maskbits[w]);
  red[tid] = c;
  __syncthreads();
  for (int off = 128; off > 0; off >>= 1) {
    if (tid < off) red[tid] += red[tid + off];
    __syncthreads();
  }
  if (tid == 0) inter[i] = red[0];
}

// ---------------------------------------------------------------------------
// 3) Argmax over 64 recalls; decide cached vs topk path.
// ---------------------------------------------------------------------------
__global__ __launch_bounds__(64) void decide_kernel(
    const unsigned* __restrict__ inter, unsigned* __restrict__ flags) {
  __shared__ unsigned v[NCACHE];
  const int t = threadIdx.x;
  v[t] = inter[t];
  __syncthreads();
  if (t == 0) {
    unsigned best = v[0], bi = 0;
    for (int i = 1; i < NCACHE; ++i) if (v[i] > best) { best = v[i]; bi = i; }
    float recall = (float)best / (float)TOPK;
    flags[0] = (recall >= 0.9f) ? 1u : 0u;
    flags[1] = bi;
  }
}

// ---------------------------------------------------------------------------
// 4) Build the masked vector s.  Topk path: s[j] = mask[j]*x[j].
//    Cached path: s zeroed here, multiplicity added by scatter kernel
//    (duplicates in cached_indices count multiple times, as in reference).
// ---------------------------------------------------------------------------
__global__ __launch_bounds__(256) void build_s_kernel(
    const float* __restrict__ x, const unsigned* __restrict__ maskbits,
    const unsigned* __restrict__ flags, float* __restrict__ s) {
  int j = blockIdx.x * 256 + threadIdx.x;
  if (j < IN_F) {
    bool use_cached = flags[0] != 0u;
    bool sel = (maskbits[j >> 5] >> (j & 31)) & 1u;
    s[j] = (!use_cached && sel) ? x[j] : 0.0f;
  }
}

__global__ __launch_bounds__(256) void scatter_s_kernel(
    const float* __restrict__ x, const long long* __restrict__ cached,
    const unsigned* __restrict__ flags, float* __restrict__ s) {
  int t = blockIdx.x * 256 + threadIdx.x;
  if (t < TOPK && flags[0] != 0u) {
    unsigned idx = (unsigned)cached[(size_t)flags[1] * TOPK + t];
    atomicAdd(&s[idx], x[idx]);   // equal addends per slot -> deterministic
  }
}

// ---------------------------------------------------------------------------
// 5) Dense GEMV out = W @ s via V_WMMA_F32_16X16X4_F32 (full f32 precision).
//    Grid: 256 row-tiles x 32 K-chunks, one wave each (8192 waves -> enough
//    parallelism to saturate 23.3 TB/s; kernel is bandwidth-bound ~170:1).
//    A = 16x4 W tile (ISA layout: lanes 0-15 hold K=0,1; lanes 16-31 K=2,3).
//    B = s chunk in column 0, zeros elsewhere; D[:,0] = 16 partial outputs.
// ---------------------------------------------------------------------------
__global__ __launch_bounds__(32) void gemv_wmma_kernel(
    const float* __restrict__ W, const float* __restrict__ s,
    float* __restrict__ partial) {
  const int tile = blockIdx.x;        // output rows [tile*16, tile*16+16)
  const int q    = blockIdx.y;        // K-chunk  [q*CHUNK, q*CHUNK+CHUNK)
  const int lane = threadIdx.x;
  const int n    = lane & 15;         // B/D column
  const int half = lane >> 4;         // K pair selector
  const int m    = lane & 15;         // A row within tile

  const float* wrow = W + (size_t)(tile * 16 + m) * IN_F + 2 * half;
  const float* sp   = s + 2 * half;

  v8f c = {};
  const int j0 = q * CHUNK;
  #pragma unroll 2
  for (int j = j0; j < j0 + CHUNK; j += 4) {
    v2f a  = *(const v2f*)(wrow + j);         // A[m][2h], A[m][2h+1]
    v2f sv = *(const v2f*)(sp + j);           // s[j+2h], s[j+2h+1]
    v2f b;
    b.x = (n == 0) ? sv.x : 0.0f;             // B[2h  ][n]
    b.y = (n == 0) ? sv.y : 0.0f;             // B[2h+1][n]
    c = __builtin_amdgcn_wmma_f32_16x16x4_f32(
        /*neg_a=*/false, a, /*neg_b=*/false, b,
        /*c_mod=*/(short)0, c, /*reuse_a=*/false, /*reuse_b=*/false);
  }

  // D[M][0] lives in lane 0 (M = r) and lane 16 (M = 8 + r), VGPR r.
  if (n == 0) {
    float* p = partial + ((size_t)tile * NCHUNK + q) * 16 + half * 8;
    #pragma unroll
    for (int r = 0; r < 8; ++r) p[r] = c[r];
  }
}

// ---------------------------------------------------------------------------
// 6) Combine K-chunk partials + bias.
// ---------------------------------------------------------------------------
__global__ __launch_bounds__(256) void combine_kernel(
    const float* __restrict__ partial, const float* __restrict__ bias,
    float* __restrict__ out) {
  int o = blockIdx.x * 256 + threadIdx.x;
  if (o < OUT_F) {
    int tile = o >> 4, r = o & 15;
    float acc = bias[o];
    #pragma unroll 4
    for (int qq = 0; qq < NCHUNK; ++qq)
      acc += partial[((size_t)tile * NCHUNK + qq) * 16 + r];
    out[o] = acc;
  }
}

// ---------------------------------------------------------------------------
extern "C" void kernel_launch(void* const* d_in, const int* in_sizes, int n_in,
                              void* d_out, int out_size, void* d_ws, size_t ws_size,
                              hipStream_t stream) {
  const float*     x      = (const float*)d_in[0];
  const float*     W      = (const float*)d_in[1];
  const float*     bias   = (const float*)d_in[2];
  const long long* cached = (const long long*)d_in[3];   // int64 indices
  float*           out    = (float*)d_out;

  // workspace layout (elements)
  unsigned* u        = (unsigned*)d_ws;
  unsigned* maskbits = u;                 // [0,   344)
  unsigned* inter    = u + 512;           // [512, 576)
  unsigned* flags    = u + 576;           // [576, 578)  {use_cached, best_idx}
  float*    s        = (float*)d_ws + 1024;    // 11008 floats
  float*    partial  = (float*)d_ws + 16384;   // 256*32*16 = 131072 floats

  select_topk_kernel<<<1, 1024, 0, stream>>>(x, maskbits);
  recall_kernel<<<NCACHE, 256, 0, stream>>>(cached, maskbits, inter);
  decide_kernel<<<1, 64, 0, stream>>>(inter, flags);
  build_s_kernel<<<(IN_F + 255) / 256, 256, 0, stream>>>(x, maskbits, flags, s);
  scatter_s_kernel<<<(TOPK + 255) / 256, 256, 0, stream>>>(x, cached, flags, s);
  gemv_wmma_kernel<<<dim3(OUT_F / 16, NCHUNK), 32, 0, stream>>>(W, s, partial);
  combine_kernel<<<(OUT_F + 255) / 256, 256, 0, stream>>>(partial, bias, out);
}